// TimeDomainModel_61873298866764
// MI455X (gfx1250) — compile-verified
//
#include <hip/hip_runtime.h>
#include <math.h>

// Euler-Maruyama scan as:
//   local(16x64) = T(16x16 lower-tri, T[i,j]=a^{i-j}) @ Bmat(16x64)  [16x v_wmma_f32_16x16x4_f32]
//   cross-block carry: wave32 parallel affine scan (5 shuffle steps)
//   y[16J+i] = beta * (local[i,J] + a^{i+1} * carry[J])

#define LENGTH 1000
#define NPAD   1024
#define NBLK   64
#define BSTR   20          // padded LDS row stride (dwords): bank-conflict-free, 16B aligned
#define DT_    0.01f

typedef __attribute__((ext_vector_type(2))) float v2f;
typedef __attribute__((ext_vector_type(4))) float v4f;
typedef __attribute__((ext_vector_type(8))) float v8f;

__launch_bounds__(32)
__global__ void TimeDomainModel_61873298866764_kernel(
    const float* __restrict__ sig_p,
    const float* __restrict__ mu_p,
    const float* __restrict__ lamb_p,
    const float* __restrict__ beta_p,
    const float* __restrict__ noise,
    float* __restrict__ out)
{
    __shared__ float lds_b[NBLK * BSTR];    // forcing vector, row-padded
    __shared__ float lds_loc[NBLK * BSTR];  // local[J][i] at J*BSTR + i
    __shared__ float lds_tail[2 * NBLK];    // [64..127] = local[15][J]
    __shared__ float lds_carry[NBLK];
    __shared__ float lds_apow[17];          // a^0 .. a^16

    const int lane = threadIdx.x;           // one wave32
    const int m    = lane & 15;
    const int hi   = lane >> 4;

    const float sigma = sig_p[0];
    const float mu    = mu_p[0];
    const float lamb  = lamb_p[0];
    const float beta  = beta_p[0];

    const float a    = fmaf(-DT_, lamb, 1.0f);   // 1 - dt*lamb
    const float smul = sigma * sqrtf(DT_);
    const float bm   = DT_ * mu;

    // ---- phase A: forcing vector + a^lane power table (square-multiply) --
    for (int t = lane; t < NPAD; t += 32) {
        float b = 0.0f;
        if (t < LENGTH) b = fmaf(smul, noise[t], bm);
        lds_b[(t >> 4) * BSTR + (t & 15)] = b;
    }
    {
        float p = 1.0f, s = a;               // p = a^lane, 5 select steps
        p = (lane & 1)  ? p * s : p; s *= s;
        p = (lane & 2)  ? p * s : p; s *= s;
        p = (lane & 4)  ? p * s : p; s *= s;
        p = (lane & 8)  ? p * s : p; s *= s;
        p = (lane & 16) ? p * s : p;
        if (lane <= 16) lds_apow[lane] = p;
    }
    __syncthreads();

    // ---- phase B: A operand (T columns) built once, pure VALU ------------
    // A chunk k: VGPR0 = T[m][4k+2hi], VGPR1 = T[m][4k+2hi+1]; T[i][j]=a^(i-j), j<=i
    const float ainv  = 1.0f / a;
    const float ainv4 = (ainv * ainv) * (ainv * ainv);
    float am = 1.0f, s2 = a;                 // a^m via 4-bit square-multiply
    am = (m & 1) ? am * s2 : am; s2 *= s2;
    am = (m & 2) ? am * s2 : am; s2 *= s2;
    am = (m & 4) ? am * s2 : am; s2 *= s2;
    am = (m & 8) ? am * s2 : am;
    const int   D  = m - 2 * hi;
    float       v  = hi ? am * (ainv * ainv) : am;   // a^D (valid when D>=0)
    v2f A[4];
    #pragma unroll
    for (int k = 0; k < 4; ++k) {
        const int d0 = D - 4 * k;
        A[k].x = (d0 >= 0) ? v        : 0.0f;
        A[k].y = (d0 >= 1) ? v * ainv : 0.0f;
        v *= ainv4;
    }

    // 4 N-tiles x 4 K-chunks of v_wmma_f32_16x16x4_f32
    for (int G = 0; G < 4; ++G) {
        const int J    = 16 * G + m;
        const int base = J * BSTR + 2 * hi;
        v8f c = {};
        #pragma unroll
        for (int k = 0; k < 4; ++k) {
            v2f B;
            B.x = lds_b[base + 4 * k];
            B.y = lds_b[base + 4 * k + 1];
            c = __builtin_amdgcn_wmma_f32_16x16x4_f32(
                    false, A[k], false, B, (short)0, c, false, false);
        }
        // C/D layout: VGPR r = row i = r + 8*hi, col J -> two b128 stores
        v4f* dst = reinterpret_cast<v4f*>(&lds_loc[J * BSTR + 8 * hi]);
        v4f lo4 = {c[0], c[1], c[2], c[3]};
        v4f hi4 = {c[4], c[5], c[6], c[7]};
        dst[0] = lo4;
        dst[1] = hi4;
        // tails: hi lanes hold local[15][J] in c[7]; unconditional store
        lds_tail[hi * NBLK + J] = c[7];
    }
    __syncthreads();

    // ---- phase C: wave32 parallel carry scan (2 blocks per lane) ---------
    // carry_{J+1} = a16*carry_J + tail_J ; lane L composes blocks 2L,2L+1
    const float a16 = lds_apow[16];
    const float a32 = a16 * a16;
    const float t0  = lds_tail[NBLK + 2 * lane];
    const float t1  = lds_tail[NBLK + 2 * lane + 1];
    float Aacc = a32;
    float Bacc = fmaf(a16, t0, t1);          // carry_{2L+2} = Aacc*carry_{2L} + Bacc
    #pragma unroll
    for (int o = 1; o < 32; o <<= 1) {
        const float Ap = __shfl_up(Aacc, o, 32);
        const float Bp = __shfl_up(Bacc, o, 32);
        if (lane >= o) {
            Bacc = fmaf(Aacc, Bp, Bacc);
            Aacc = Aacc * Ap;
        }
    }
    float cpre = __shfl_up(Bacc, 1, 32);     // carry_{2L}
    if (lane == 0) cpre = 0.0f;
    const float cmid = fmaf(a16, cpre, t0);  // carry_{2L+1}
    lds_carry[2 * lane]     = cpre;
    lds_carry[2 * lane + 1] = cmid;
    __syncthreads();

    // ---- phase D: outputs; i = lane&15 is loop-invariant -----------------
    const float api1 = lds_apow[m + 1];      // a^{i+1}
    for (int t = lane; t < LENGTH; t += 32) {
        const int J = t >> 4;
        const float y = fmaf(api1, lds_carry[J], lds_loc[J * BSTR + m]);
        out[t] = beta * y;
    }
}

extern "C" void kernel_launch(void* const* d_in, const int* in_sizes, int n_in,
                              void* d_out, int out_size, void* d_ws, size_t ws_size,
                              hipStream_t stream) {
    (void)in_sizes; (void)n_in; (void)out_size; (void)d_ws; (void)ws_size;
    const float* sigma = (const float*)d_in[0];
    const float* mu    = (const float*)d_in[1];
    const float* lamb  = (const float*)d_in[2];
    const float* beta  = (const float*)d_in[3];
    const float* noise = (const float*)d_in[4];
    float* out = (float*)d_out;
    TimeDomainModel_61873298866764_kernel<<<1, 32, 0, stream>>>(
        sigma, mu, lamb, beta, noise, out);
}